// WaveNetStyleModel_55216099557519
// MI455X (gfx1250) — compile-verified
//
#include <hip/hip_runtime.h>
#include <math.h>

// Problem constants (from reference)
constexpr int kB   = 32;    // batch
constexpr int kT   = 4096;  // seq len
constexpr int kFIN = 256;   // input features
constexpr int kC   = 64;    // residual channels
constexpr int kS   = 128;   // skip channels
constexpr int kL   = 8;     // layers
constexpr int kNT  = 424;   // targets
constexpr int kRF  = 256;   // receptive field (1+2+...+128 + 1)

typedef __attribute__((ext_vector_type(2))) float v2f;
typedef __attribute__((ext_vector_type(8))) float v8f;

// D = A(16x4) * B(4x16) + C, fp32 exact.  8-arg form per probe doc:
// (neg_a, A, neg_b, B, c_mod, C, reuse_a, reuse_b)
__device__ __forceinline__ v8f wmma4(v2f a, v2f b, v8f c) {
  return __builtin_amdgcn_wmma_f32_16x16x4_f32(false, a, false, b, (short)0, c,
                                               false, false);
}

// -------------------------------------------------------------------------
// Kernel 1: h0[b][u][o] = W_in @ x  for the last 256 timesteps only.
// grid = (16 M-tiles, 32 batches), block = 128 (4 waves, one N-tile each).
// -------------------------------------------------------------------------
__global__ __launch_bounds__(128) void k_input_gemm(
    const float* __restrict__ x, const float* __restrict__ Win,
    const float* __restrict__ bin, float* __restrict__ h0) {
  const int mt   = blockIdx.x;          // time tile 0..15
  const int b    = blockIdx.y;          // batch
  const int lane = threadIdx.x & 31;
  const int nt0  = threadIdx.x >> 5;    // output-channel tile 0..3
  const int l15  = lane & 15;
  const int lh   = lane >> 4;

  const long trow = (long)b * kT + (kT - kRF) + mt * 16 + l15;
  const float* xrow = x + trow * kFIN;                 // A row (M = time)
  const float* wrow = Win + (nt0 * 16 + l15) * kFIN;   // B col source (N = o)

  v8f c = {};
#pragma unroll 8
  for (int ks = 0; ks < kFIN / 4; ++ks) {
    const int kc = ks * 4 + 2 * lh;                    // K pair for this half
    v2f a  = *reinterpret_cast<const v2f*>(xrow + kc);
    v2f bb = *reinterpret_cast<const v2f*>(wrow + kc);
    c = wmma4(a, bb, c);
  }
  const float bias = bin[nt0 * 16 + l15];
  float* hb = h0 + (long)(b * kRF + mt * 16) * kC;
#pragma unroll
  for (int r = 0; r < 8; ++r) {
    const int row = r + lh * 8;                        // C/D row layout
    hb[row * kC + nt0 * 16 + l15] = c[r] + bias;
  }
}

// -------------------------------------------------------------------------
// Kernel 2: dependency-cone WaveNet layers + head. 1 workgroup per batch.
// LDS: bufA[256][64] | bufB[128][64] | gated[128][64] | skip[128] | o1[128]
// Position index k within a layer means time offset u = 255 - k*2^layer.
// -------------------------------------------------------------------------
extern __shared__ float smem[];

__global__ __launch_bounds__(512) void k_cone(
    const float* __restrict__ h0,
    const float* __restrict__ Wf, const float* __restrict__ bfp,
    const float* __restrict__ Wg, const float* __restrict__ bgp,
    const float* __restrict__ Wsk, const float* __restrict__ bsk,
    const float* __restrict__ Wr, const float* __restrict__ brp,
    const float* __restrict__ W1, const float* __restrict__ b1p,
    const float* __restrict__ W2, const float* __restrict__ b2p,
    float* __restrict__ out) {
  float* bufA    = smem;                       // 256*64
  float* bufB    = bufA + kRF * kC;            // 128*64
  float* gated   = bufB + (kRF / 2) * kC;      // 128*64
  float* skipAcc = gated + (kRF / 2) * kC;     // 128
  float* o1buf   = skipAcc + kS;               // 128

  const int b    = blockIdx.x;
  const int tid  = threadIdx.x;
  const int lane = tid & 31;
  const int wave = tid >> 5;
  const int NW   = blockDim.x >> 5;            // 16 waves
  const int l15  = lane & 15;
  const int lh   = lane >> 4;

  // Load h0 reversed: bufA[k][c] = h0[b][255-k][c]  (k=0 is t=T-1)
  for (int idx = tid; idx < kRF * kC; idx += blockDim.x) {
    const int k = idx >> 6, cc = idx & 63;
    bufA[idx] = h0[((long)b * kRF + (kRF - 1 - k)) * kC + cc];
  }
  if (tid < kS) skipAcc[tid] = 0.f;
  __syncthreads();

  float* inb  = bufA;
  float* outb = bufB;

  for (int layer = 0; layer < kL; ++layer) {
    const int Pin  = kRF >> layer;
    const int Pout = Pin >> 1;
    const int Mt   = (Pout + 15) >> 4;         // >= 1 (padded tiles clamp reads)
    const int kmax = Pin - 1;

    // ---- Stage A: f & g GEMMs (K = 2 taps x 64 ch) + gating, per wave ----
    const float* Wfl = Wf + layer * kC * kC * 2;
    const float* Wgl = Wg + layer * kC * kC * 2;
    for (int task = wave; task < Mt * 4; task += NW) {
      const int mt  = task >> 2;
      const int nt0 = task & 3;
      const int o   = nt0 * 16 + l15;          // B-lane column (output ch)
      v8f cf = {}, cg = {};
#pragma unroll
      for (int tap = 0; tap < 2; ++tap) {
        // weight idx 0 pairs h[t-d] (row 2m+1); weight idx 1 pairs h[t] (row 2m)
        const int roff = (tap == 0) ? 1 : 0;
        const int widx = (tap == 0) ? 0 : 1;
        int arow = 2 * (mt * 16 + l15) + roff;
        if (arow > kmax) arow = kmax;          // clamp: pad rows read valid data
        const float* arp = inb + arow * kC;
#pragma unroll 4
        for (int ks = 0; ks < 16; ++ks) {
          const int kc = ks * 4 + 2 * lh;
          v2f a = *reinterpret_cast<const v2f*>(arp + kc);   // LDS (ds_load_b64)
          const int wb = (o * kC + kc) * 2 + widx;           // [o][c][tap]
          v2f bfv, bgv;
          bfv.x = Wfl[wb]; bfv.y = Wfl[wb + 2];
          bgv.x = Wgl[wb]; bgv.y = Wgl[wb + 2];
          cf = wmma4(a, bfv, cf);
          cg = wmma4(a, bgv, cg);
        }
      }
      const float bfo = bfp[layer * kC + o];
      const float bgo = bgp[layer * kC + o];
#pragma unroll
      for (int r = 0; r < 8; ++r) {
        const int m = mt * 16 + lh * 8 + r;    // C/D row layout
        const float fv = tanhf(cf[r] + bfo);
        const float gv = 1.f / (1.f + expf(-(cg[r] + bgo)));
        gated[m * kC + o] = fv * gv;
      }
    }
    __syncthreads();

    // ---- Stage B: residual h_new = h[2m] + br + Wr @ gated (WMMA) ----
    if (layer < kL - 1) {                      // last layer's h is unused
      const float* Wrl = Wr + layer * kC * kC;
      for (int task = wave; task < Mt * 4; task += NW) {
        const int mt  = task >> 2;
        const int nt0 = task & 3;
        const int o   = nt0 * 16 + l15;
        const float bro = brp[layer * kC + o];
        v8f c;
#pragma unroll
        for (int r = 0; r < 8; ++r) {          // init accumulator with h + br
          const int m = mt * 16 + lh * 8 + r;
          int arow = 2 * m; if (arow > kmax) arow = kmax;
          c[r] = inb[arow * kC + o] + bro;
        }
        const float* grp = gated + (mt * 16 + l15) * kC;     // A rows
        const float* wrp = Wrl + o * kC;                     // B cols
#pragma unroll 4
        for (int ks = 0; ks < 16; ++ks) {
          const int kc = ks * 4 + 2 * lh;
          v2f a  = *reinterpret_cast<const v2f*>(grp + kc);
          v2f bb = *reinterpret_cast<const v2f*>(wrp + kc);
          c = wmma4(a, bb, c);
        }
#pragma unroll
        for (int r = 0; r < 8; ++r) {
          const int m = mt * 16 + lh * 8 + r;
          outb[m * kC + o] = c[r];
        }
      }
    }
    __syncthreads();

    // ---- skip contribution at t = T-1 (gated row 0) ----
    if (tid < kS) {
      const float* wsp = Wsk + (layer * kS + tid) * kC;
      float acc = bsk[layer * kS + tid];
#pragma unroll 8
      for (int cc = 0; cc < kC; ++cc) acc += wsp[cc] * gated[cc];
      skipAcc[tid] += acc;
    }
    __syncthreads();

    float* t = inb; inb = outb; outb = t;      // ping-pong
  }

  // ---- head: relu -> W1 -> relu -> W2, only at final timestep ----
  if (tid < kS) {
    const float* w1p = W1 + tid * kS;
    float acc = b1p[tid];
    for (int s = 0; s < kS; ++s) {
      float sv = skipAcc[s]; sv = sv > 0.f ? sv : 0.f;
      acc += w1p[s] * sv;
    }
    o1buf[tid] = acc > 0.f ? acc : 0.f;
  }
  __syncthreads();
  if (tid < kNT) {
    const float* w2p = W2 + tid * kS;
    float acc = b2p[tid];
    for (int s = 0; s < kS; ++s) acc += w2p[s] * o1buf[s];
    out[b * kNT + tid] = acc;
  }
}

// -------------------------------------------------------------------------
extern "C" void kernel_launch(void* const* d_in, const int* in_sizes, int n_in,
                              void* d_out, int out_size, void* d_ws,
                              size_t ws_size, hipStream_t stream) {
  const float* x   = (const float*)d_in[0];
  const float* Win = (const float*)d_in[1];
  const float* bin = (const float*)d_in[2];
  const float* Wf  = (const float*)d_in[3];
  const float* bf  = (const float*)d_in[4];
  const float* Wg  = (const float*)d_in[5];
  const float* bg  = (const float*)d_in[6];
  const float* Wsk = (const float*)d_in[7];
  const float* bsk = (const float*)d_in[8];
  const float* Wr  = (const float*)d_in[9];
  const float* br  = (const float*)d_in[10];
  const float* W1  = (const float*)d_in[11];
  const float* b1  = (const float*)d_in[12];
  const float* W2  = (const float*)d_in[13];
  const float* b2  = (const float*)d_in[14];
  float* out = (float*)d_out;
  float* h0  = (float*)d_ws;   // kB * kRF * kC floats = 2 MB

  k_input_gemm<<<dim3(16, kB), 128, 0, stream>>>(x, Win, bin, h0);

  const size_t shbytes =
      (size_t)(kRF * kC + (kRF / 2) * kC * 2 + 2 * kS) * sizeof(float);
  k_cone<<<dim3(kB), 512, shbytes, stream>>>(h0, Wf, bf, Wg, bg, Wsk, bsk, Wr,
                                             br, W1, b1, W2, b2, out);
}